// DiceLoss_11038065951148
// MI455X (gfx1250) — compile-verified
//
#include <hip/hip_runtime.h>
#include <hip/hip_bf16.h>

// ---------------------------------------------------------------------------
// DiceLoss on integer label maps, N = 16.7M, C = 4.
// Math collapses: only diagonal of cm accumulates -> fn = fp = 0, so we only
// need tp[c] = count(pred==target==c) for c in {1,2,3}.
// Phase 1: bandwidth-bound grid-stride count (int4 / global_load_b128).
// Phase 2: per-wave cross-lane reduction via V_WMMA_F32_16X16X4_F32.
// Phase 3: 1-thread epilogue computing balance * (1 - dice^0.75).
// ---------------------------------------------------------------------------

typedef float v2f __attribute__((ext_vector_type(2)));
typedef float v8f __attribute__((ext_vector_type(8)));

__global__ void dice_zero_ws(unsigned* __restrict__ ws) {
    if (threadIdx.x < 4) ws[threadIdx.x] = 0u;
}

__global__ __launch_bounds__(256) void dice_count(const int4* __restrict__ pred,
                                                  const int4* __restrict__ tgt,
                                                  unsigned* __restrict__ ws,
                                                  int n4) {
    const int stride = gridDim.x * blockDim.x;
    int c1 = 0, c2 = 0, c3 = 0;

    for (int i = blockIdx.x * blockDim.x + threadIdx.x; i < n4; i += stride) {
        // Prefetch next grid-stride iteration (speculative; OOB is dropped).
        __builtin_prefetch(&tgt[i + stride], 0, 3);
        __builtin_prefetch(&pred[i + stride], 0, 3);

        const int4 t = tgt[i];
        const int4 p = pred[i];
        c1 += (t.x == p.x) & (t.x == 1);
        c2 += (t.x == p.x) & (t.x == 2);
        c3 += (t.x == p.x) & (t.x == 3);
        c1 += (t.y == p.y) & (t.y == 1);
        c2 += (t.y == p.y) & (t.y == 2);
        c3 += (t.y == p.y) & (t.y == 3);
        c1 += (t.z == p.z) & (t.z == 1);
        c2 += (t.z == p.z) & (t.z == 2);
        c3 += (t.z == p.z) & (t.z == 3);
        c1 += (t.w == p.w) & (t.w == 1);
        c2 += (t.w == p.w) & (t.w == 2);
        c3 += (t.w == p.w) & (t.w == 3);
    }

    // ---- Wave32 cross-lane reduction via WMMA (EXEC is all-1s here) ------
    // A (16x4 f32): VGPR0 = K0 (lanes 0-15) / K2 (lanes 16-31),
    //               VGPR1 = K1 / K3.  A.x=c1, A.y=c2 gives
    //               A[m][0]=c1(m), A[m][2]=c1(m+16), A[m][1]=c2(m), A[m][3]=c2(m+16).
    // B (4x16 f32): VGPR0 = row0/row2, VGPR1 = row1/row3 (C/D-style striping).
    //   Choose B[0][n]=B[2][n]=(n<8), B[1][n]=B[3][n]=(n>=8):
    //   -> D[m][n<8]  = c1(m)+c1(m+16)
    //      D[m][n>=8] = c2(m)+c2(m+16)
    const unsigned lane = threadIdx.x & 31u;
    const float bsel0 = ((lane & 15u) < 8u) ? 1.0f : 0.0f;

    v2f A1;  A1.x = (float)c1;  A1.y = (float)c2;
    v2f A2;  A2.x = (float)c3;  A2.y = 0.0f;
    v2f B;   B.x  = bsel0;      B.y  = 1.0f - bsel0;

    v8f acc1 = {};
    v8f acc2 = {};
    acc1 = __builtin_amdgcn_wmma_f32_16x16x4_f32(false, A1, false, B,
                                                 (short)0, acc1, false, false);
    acc2 = __builtin_amdgcn_wmma_f32_16x16x4_f32(false, A2, false, B,
                                                 (short)0, acc2, false, false);

    // Sum D rows held in the 8 accumulator VGPRs (in-lane adds):
    // lane n  (n<16)  -> sum over m=0..7  of D[m][n]
    // lane n+16       -> sum over m=8..15 of D[m][n]
    float r1 = acc1[0] + acc1[1] + acc1[2] + acc1[3] +
               acc1[4] + acc1[5] + acc1[6] + acc1[7];
    float r2 = acc2[0] + acc2[1] + acc2[2] + acc2[3] +
               acc2[4] + acc2[5] + acc2[6] + acc2[7];

    // Fold the two half-wave row groups together.
    r1 += __shfl_xor(r1, 16, 32);
    r2 += __shfl_xor(r2, 16, 32);

    // lane 0  : r1 = wave total c1, r2 = wave total c3
    // lane 8  : r1 = wave total c2
    if (lane == 0u) {
        atomicAdd(&ws[1], (unsigned)(r1 + 0.5f));
        atomicAdd(&ws[3], (unsigned)(r2 + 0.5f));
    } else if (lane == 8u) {
        atomicAdd(&ws[2], (unsigned)(r1 + 0.5f));
    }
}

__global__ void dice_final(const unsigned* __restrict__ ws,
                           const int* __restrict__ pred,
                           const int* __restrict__ tgt,
                           const int* __restrict__ numc,
                           const float* __restrict__ balance,
                           float* __restrict__ out,
                           int n, int n_vec) {
    float tp[4];
    tp[0] = 0.0f;
    tp[1] = (float)ws[1];
    tp[2] = (float)ws[2];
    tp[3] = (float)ws[3];

    // Scalar tail (n % 4 elements) — zero for N = 16.7M but kept for safety.
    for (int i = n_vec; i < n; ++i) {
        const int t = tgt[i];
        if (t == pred[i] && t >= 1 && t <= 3) tp[t] += 1.0f;
    }

    const int C = *numc;
    float dice = 0.0f;
    for (int c = 1; c < 4; ++c) {
        dice += (2.0f * tp[c]) / (2.0f * tp[c] + 1e-6f);
    }
    dice /= (float)(C - 1);

    out[0] = balance[0] * (1.0f - powf(dice, 0.75f));
}

extern "C" void kernel_launch(void* const* d_in, const int* in_sizes, int n_in,
                              void* d_out, int out_size, void* d_ws, size_t ws_size,
                              hipStream_t stream) {
    const int*   pred    = (const int*)d_in[0];
    const int*   tgt     = (const int*)d_in[1];
    const int*   numc    = (const int*)d_in[2];
    const float* balance = (const float*)d_in[3];
    float*       out     = (float*)d_out;
    unsigned*    ws      = (unsigned*)d_ws;

    const int n  = in_sizes[0];
    const int n4 = n / 4;

    dice_zero_ws<<<1, 32, 0, stream>>>(ws);

    const int threads = 256;
    int blocks = 2048;                              // ~8 int4 per thread at N=16.7M
    const int needed = (n4 + threads - 1) / threads;
    if (blocks > needed) blocks = needed;
    if (blocks < 1) blocks = 1;

    dice_count<<<blocks, threads, 0, stream>>>((const int4*)pred, (const int4*)tgt,
                                               ws, n4);

    dice_final<<<1, 1, 0, stream>>>(ws, pred, tgt, numc, balance, out, n, n4 * 4);
}